// GCN_88648124991285
// MI455X (gfx1250) — compile-verified
//
#include <hip/hip_runtime.h>
#include <hip/hip_bf16.h>

#define N_NODES 100000
#define IN_DIM  116
#define HIDDEN  256

typedef float v2f __attribute__((ext_vector_type(2)));
typedef float v8f __attribute__((ext_vector_type(8)));

// ---------------------------------------------------------------------------
// deg[i] = 1.0 (self loop)
__global__ void gcn_deg_init(float* __restrict__ deg) {
    int i = blockIdx.x * 256 + threadIdx.x;
    if (i < N_NODES) deg[i] = 1.0f;
}

// deg[dst[e]] += 1
__global__ void gcn_deg_accum(const int* __restrict__ dst, float* __restrict__ deg, int nE) {
    int e = blockIdx.x * 256 + threadIdx.x;
    if (e < nE) atomicAdd(&deg[dst[e]], 1.0f);
}

// dinv[i] = deg>0 ? rsqrt(deg) : 0
__global__ void gcn_dinv(const float* __restrict__ deg, float* __restrict__ dinv) {
    int i = blockIdx.x * 256 + threadIdx.x;
    if (i < N_NODES) {
        float d = deg[i];
        dinv[i] = (d > 0.0f) ? rsqrtf(d) : 0.0f;
    }
}

// ---------------------------------------------------------------------------
// hs = (x @ W) * dinv[row]   -- fp32 WMMA 16x16x4, one 16-row M-tile per block,
// 4 waves each own a 64-wide N-strip (4 accumulators). K=116 = 29 steps of 4.
__global__ __launch_bounds__(128) void gcn_gemm_scale(
    const float* __restrict__ x, const float* __restrict__ W,
    const float* __restrict__ dinv, float* __restrict__ hs)
{
    __shared__ float Atile[16][IN_DIM + 5];   // stride 121 (odd) -> conflict-free

    const int tid     = threadIdx.x;          // 0..127
    const int wave    = tid >> 5;             // 0..3
    const int lane    = tid & 31;
    const int halfSel = lane >> 4;            // 0: lanes 0-15, 1: lanes 16-31
    const int l15     = lane & 15;
    const int rowBase = blockIdx.x * 16;      // 100000/16 = 6250 blocks exactly

    // Cooperative, coalesced load of the 16x116 A tile into LDS.
    for (int i = tid; i < 16 * IN_DIM; i += 128) {
        int r = i / IN_DIM, c = i % IN_DIM;
        Atile[r][c] = x[(size_t)(rowBase + r) * IN_DIM + c];
    }
    __syncthreads();

    const int colBase = wave * 64;

    v8f acc0 = {}, acc1 = {}, acc2 = {}, acc3 = {};

    for (int k0 = 0; k0 < IN_DIM; k0 += 4) {
        const int kA = k0 + 2 * halfSel;      // this half-wave's K pair

        // A fragment (16x4): lane row = l15, two consecutive K values.
        v2f a;
        a.x = Atile[l15][kA];
        a.y = Atile[l15][kA + 1];

        // B fragments (4x16 each) straight from global W[116][256] (L2-hot).
        const float* Wp = W + (size_t)kA * HIDDEN + colBase + l15;
        v2f b0, b1, b2, b3;
        b0.x = Wp[0];      b0.y = Wp[HIDDEN];
        b1.x = Wp[16];     b1.y = Wp[HIDDEN + 16];
        b2.x = Wp[32];     b2.y = Wp[HIDDEN + 32];
        b3.x = Wp[48];     b3.y = Wp[HIDDEN + 48];

        acc0 = __builtin_amdgcn_wmma_f32_16x16x4_f32(false, a, false, b0, (short)0, acc0, false, false);
        acc1 = __builtin_amdgcn_wmma_f32_16x16x4_f32(false, a, false, b1, (short)0, acc1, false, false);
        acc2 = __builtin_amdgcn_wmma_f32_16x16x4_f32(false, a, false, b2, (short)0, acc2, false, false);
        acc3 = __builtin_amdgcn_wmma_f32_16x16x4_f32(false, a, false, b3, (short)0, acc3, false, false);
    }

    // Epilogue: D VGPR r of this lane holds row (r + 8*halfSel), col = l15 (+16k).
    for (int r = 0; r < 8; ++r) {
        const int row = rowBase + r + 8 * halfSel;
        const float s = dinv[row];
        float* p = hs + (size_t)row * HIDDEN + colBase + l15;
        p[0]  = acc0[r] * s;
        p[16] = acc1[r] * s;
        p[32] = acc2[r] * s;
        p[48] = acc3[r] * s;
    }
}

// ---------------------------------------------------------------------------
// agg[dst[e]] += hs[src[e]]  -- 64 threads per edge, float4 gather + 4 atomics.
__global__ __launch_bounds__(256) void gcn_scatter(
    const int* __restrict__ src, const int* __restrict__ dst,
    const float* __restrict__ hs, float* __restrict__ agg, int nE)
{
    long long t = (long long)blockIdx.x * 256 + threadIdx.x;
    if (t >= (long long)nE * 64) return;
    const int e = (int)(t >> 6);
    const int c = ((int)t & 63) << 2;

    const int s = src[e];
    const int d = dst[e];
    const float4 v = *(const float4*)(hs + (size_t)s * HIDDEN + c);
    float* p = agg + (size_t)d * HIDDEN + c;
    atomicAdd(p + 0, v.x);
    atomicAdd(p + 1, v.y);
    atomicAdd(p + 2, v.z);
    atomicAdd(p + 3, v.w);
}

// ---------------------------------------------------------------------------
// out = relu(dinv[i] * agg + b[c])   (in place on d_out)
__global__ __launch_bounds__(256) void gcn_finalize(
    float* __restrict__ out, const float* __restrict__ dinv,
    const float* __restrict__ bias)
{
    size_t t = (size_t)blockIdx.x * 256 + threadIdx.x;   // N*H = 25.6M exact
    const int i = (int)(t >> 8);
    const int c = (int)(t & 255);
    float v = out[t] * dinv[i] + bias[c];
    out[t] = fmaxf(v, 0.0f);
}

// ---------------------------------------------------------------------------
extern "C" void kernel_launch(void* const* d_in, const int* in_sizes, int n_in,
                              void* d_out, int out_size, void* d_ws, size_t ws_size,
                              hipStream_t stream)
{
    const float* x  = (const float*)d_in[0];
    const int*   ei = (const int*)d_in[1];
    const float* W  = (const float*)d_in[2];
    const float* b  = (const float*)d_in[3];

    const int nE = in_sizes[1] / 2;          // 3,200,000
    const int* src = ei;
    const int* dst = ei + nE;

    float* out = (float*)d_out;

    char*  ws   = (char*)d_ws;
    float* hs   = (float*)ws;                                       // N*H f32
    float* deg  = (float*)(ws + (size_t)N_NODES * HIDDEN * 4);      // N f32
    float* dinv = deg + N_NODES;                                    // N f32

    const int NB = (N_NODES + 255) / 256;

    gcn_deg_init <<<NB, 256, 0, stream>>>(deg);
    gcn_deg_accum<<<(nE + 255) / 256, 256, 0, stream>>>(dst, deg, nE);
    gcn_dinv     <<<NB, 256, 0, stream>>>(deg, dinv);

    gcn_gemm_scale<<<N_NODES / 16, 128, 0, stream>>>(x, W, dinv, hs);

    // Self-loop term: agg starts as hs.
    hipMemcpyAsync(out, hs, (size_t)N_NODES * HIDDEN * sizeof(float),
                   hipMemcpyDeviceToDevice, stream);

    const long long scatterThreads = (long long)nE * 64;
    const int scatterBlocks = (int)((scatterThreads + 255) / 256);
    gcn_scatter<<<scatterBlocks, 256, 0, stream>>>(src, dst, hs, out, nE);

    gcn_finalize<<<(N_NODES * HIDDEN) / 256, 256, 0, stream>>>(out, dinv, b);
}